// GCN_Fusion_21680994910212
// MI455X (gfx1250) — compile-verified
//
#include <hip/hip_runtime.h>
#include <math.h>

#define BS   2
#define V    2048
#define DIMC 128
#define NLK  10
#define NGK  100
#define KSEL (NGK + 1)          // self + 100
#define ROWS (BS * V)           // 4096
#define NOUT 256                // (SUPPORT+1)*DIM
#define EPS_NORM 1e-12f
#define EPS_BN   1e-5f

typedef __attribute__((ext_vector_type(2))) float v2f;
typedef __attribute__((ext_vector_type(8))) float v8f;

// ---------------------------------------------------------------------------
// KNN: one block per query point. Distances to all 2048 points in LDS, then
// 101 rounds of tie-break-stable argmin (lower index wins on equal value,
// matching jax.lax.top_k). Round 0 selects self (dist exactly 0) -> dropped.
// ---------------------------------------------------------------------------
__global__ __launch_bounds__(256)
void knn_kernel(const float* __restrict__ verts, int* __restrict__ idxg)
{
    __shared__ float dist[V];
    __shared__ float redv[256];
    __shared__ int   redi[256];

    const int q = blockIdx.x;          // 0..ROWS-1
    const int b = q >> 11;             // q / V
    const int i = q & (V - 1);
    const int t = threadIdx.x;

    const float* vb = verts + (size_t)b * V * 3;
    const float qx = vb[i * 3 + 0];
    const float qy = vb[i * 3 + 1];
    const float qz = vb[i * 3 + 2];
    const float qq = qx * qx + qy * qy + qz * qz;

    for (int j = t; j < V; j += 256) {
        const float x = vb[j * 3 + 0];
        const float y = vb[j * 3 + 1];
        const float z = vb[j * 3 + 2];
        const float jj = x * x + y * y + z * z;
        const float in = qx * x + qy * y + qz * z;
        dist[j] = qq - 2.0f * in + jj;   // same formula as reference
    }
    __syncthreads();

    for (int s = 0; s < KSEL; ++s) {
        float bv = INFINITY;
        int   bi = V;
        for (int j = t; j < V; j += 256) {
            const float v = dist[j];
            if (v < bv) { bv = v; bi = j; }   // strict < keeps earliest index
        }
        redv[t] = bv; redi[t] = bi;
        __syncthreads();
        for (int off = 128; off > 0; off >>= 1) {
            if (t < off) {
                const float v2 = redv[t + off];
                const int   i2 = redi[t + off];
                if (v2 < redv[t] || (v2 == redv[t] && i2 < redi[t])) {
                    redv[t] = v2; redi[t] = i2;
                }
            }
            __syncthreads();
        }
        if (t == 0) {
            const int sel = redi[0];
            if (s > 0) idxg[(size_t)q * NGK + (s - 1)] = sel;
            dist[sel] = INFINITY;
        }
        __syncthreads();
    }
}

// ---------------------------------------------------------------------------
// fo = A(ROWS x 128) @ W(128 x 256) + bias, fp32 via V_WMMA_F32_16X16X4_F32.
// One wave per 16x16 output tile; 32 chained WMMAs over K=128.
// A 16x4 layout:  lane L holds A[L&15][kb], A[L&15][kb+1], kb = (L>=16)?2:0
// B 4x16 layout:  lane L holds B[kb][L&15], B[kb+1][L&15]
// C/D 16x16:      vgpr r, lane L -> C[r + ((L>=16)?8:0)][L&15]
// ---------------------------------------------------------------------------
__global__ __launch_bounds__(256)
void gemm_bias_wmma(const float* __restrict__ A,
                    const float* __restrict__ W,
                    const float* __restrict__ bias,
                    float* __restrict__ C)
{
    const int wave = blockIdx.x * 8 + (threadIdx.x >> 5);
    const int lane = threadIdx.x & 31;
    const int mt = wave >> 4;           // 0..255  (M tile)
    const int nt = wave & 15;           // 0..15   (N tile)
    const int rc = lane & 15;           // row of A / col of B,C
    const int kb = (lane >> 4) << 1;    // 0 or 2
    const int hi = (lane >> 4) << 3;    // 0 or 8 (C row offset)

    const float* Arow = A + (size_t)(mt * 16 + rc) * DIMC;
    const float* Wcol = W + nt * 16 + rc;
    const float  bv   = bias[nt * 16 + rc];

    v8f acc;
#pragma unroll
    for (int r = 0; r < 8; ++r) acc[r] = bv;

#pragma unroll
    for (int k = 0; k < DIMC; k += 4) {
        v2f a, b;
        a.x = Arow[k + kb];
        a.y = Arow[k + kb + 1];
        b.x = Wcol[(size_t)(k + kb) * NOUT];
        b.y = Wcol[(size_t)(k + kb + 1) * NOUT];
        acc = __builtin_amdgcn_wmma_f32_16x16x4_f32(
            false, a, false, b, (short)0, acc, false, false);
    }

    float* Crow = C + (size_t)(mt * 16 + hi) * NOUT + nt * 16 + rc;
#pragma unroll
    for (int r = 0; r < 8; ++r)
        Crow[(size_t)r * NOUT] = acc[r];
}

// ---------------------------------------------------------------------------
// Fused theta / gather / max aggregation: block per point, thread per channel.
// out[q,d] = fo[q,d] + max_n relu(dot(unit(v[j]-v[i]), Dn[:,d])) * fo[j,128+d]
// ---------------------------------------------------------------------------
__global__ __launch_bounds__(DIMC)
void conv_kernel(const int* __restrict__ nbrs, int nn,
                 const float* __restrict__ verts,
                 const float* __restrict__ fo,
                 const float* __restrict__ D,
                 float* __restrict__ out)
{
    const int q = blockIdx.x;
    const int b = q >> 11;
    const int i = q & (V - 1);
    const int d = threadIdx.x;

    const float* vb = verts + (size_t)b * V * 3;
    const float cx = vb[i * 3 + 0];
    const float cy = vb[i * 3 + 1];
    const float cz = vb[i * 3 + 2];

    float d0 = D[d], d1 = D[DIMC + d], d2 = D[2 * DIMC + d];
    const float dnorm = sqrtf(d0 * d0 + d1 * d1 + d2 * d2);
    const float dinv  = 1.0f / fmaxf(dnorm, EPS_NORM);
    d0 *= dinv; d1 *= dinv; d2 *= dinv;

    const int* row = nbrs + (size_t)q * NGK;
    float acc = -INFINITY;
    for (int n = 0; n < nn; ++n) {
        const int j = row[n];                       // uniform -> scalar load
        const float nx = vb[j * 3 + 0] - cx;
        const float ny = vb[j * 3 + 1] - cy;
        const float nz = vb[j * 3 + 2] - cz;
        const float nrm = sqrtf(nx * nx + ny * ny + nz * nz);
        const float rin = 1.0f / fmaxf(nrm, EPS_NORM);
        const float theta = fmaxf((nx * d0 + ny * d1 + nz * d2) * rin, 0.0f);
        const float sup = fo[(size_t)(b * V + j) * NOUT + DIMC + d];
        acc = fmaxf(acc, theta * sup);
    }
    out[(size_t)q * DIMC + d] = fo[(size_t)q * NOUT + d] + acc;
}

// ---------------------------------------------------------------------------
// BatchNorm(axis=(0,1)) + ReLU. One block per channel.
// ---------------------------------------------------------------------------
__global__ __launch_bounds__(256)
void bn_relu_kernel(const float* __restrict__ x,
                    const float* __restrict__ gamma,
                    const float* __restrict__ beta,
                    float* __restrict__ out, int out_stride, int out_off)
{
    __shared__ float s1[256];
    __shared__ float s2[256];
    const int d = blockIdx.x;
    const int t = threadIdx.x;

    float sum = 0.0f, ssq = 0.0f;
    for (int r = t; r < ROWS; r += 256) {
        const float v = x[(size_t)r * DIMC + d];
        sum += v; ssq += v * v;
    }
    s1[t] = sum; s2[t] = ssq;
    __syncthreads();
    for (int off = 128; off > 0; off >>= 1) {
        if (t < off) { s1[t] += s1[t + off]; s2[t] += s2[t + off]; }
        __syncthreads();
    }
    const float mean  = s1[0] * (1.0f / ROWS);
    const float var   = fmaxf(s2[0] * (1.0f / ROWS) - mean * mean, 0.0f);
    const float scale = rsqrtf(var + EPS_BN) * gamma[d];
    const float shift = beta[d];
    for (int r = t; r < ROWS; r += 256) {
        const float v = (x[(size_t)r * DIMC + d] - mean) * scale + shift;
        out[(size_t)r * out_stride + out_off + d] = fmaxf(v, 0.0f);
    }
}

// ---------------------------------------------------------------------------
extern "C" void kernel_launch(void* const* d_in, const int* in_sizes, int n_in,
                              void* d_out, int out_size, void* d_ws, size_t ws_size,
                              hipStream_t stream)
{
    const float* vertices = (const float*)d_in[0];
    const float* feats    = (const float*)d_in[1];
    const float* W_l   = (const float*)d_in[2];
    const float* b_l   = (const float*)d_in[3];
    const float* D_l   = (const float*)d_in[4];
    const float* g_l   = (const float*)d_in[5];
    const float* be_l  = (const float*)d_in[6];
    const float* W_g0  = (const float*)d_in[7];
    const float* b_g0  = (const float*)d_in[8];
    const float* D_g0  = (const float*)d_in[9];
    const float* g_g0  = (const float*)d_in[10];
    const float* be_g0 = (const float*)d_in[11];
    const float* W_g1  = (const float*)d_in[12];
    const float* b_g1  = (const float*)d_in[13];
    const float* D_g1  = (const float*)d_in[14];
    const float* g_g1  = (const float*)d_in[15];
    const float* be_g1 = (const float*)d_in[16];
    float* out = (float*)d_out;

    // workspace carve-up (~9.6 MB)
    char* ws = (char*)d_ws;
    int*   idxg = (int*)ws;            ws += sizeof(int)   * (size_t)ROWS * NGK;
    float* fo   = (float*)ws;          ws += sizeof(float) * (size_t)ROWS * NOUT;
    float* pre  = (float*)ws;          ws += sizeof(float) * (size_t)ROWS * DIMC;
    float* fmg  = (float*)ws;

    const int gemm_blocks = (ROWS / 16) * (NOUT / 16) / 8;   // 512

    // 1) shared 100-NN (first 10 columns == idx_l)
    knn_kernel<<<ROWS, 256, 0, stream>>>(vertices, idxg);

    // 2) local branch
    gemm_bias_wmma<<<gemm_blocks, 256, 0, stream>>>(feats, W_l, b_l, fo);
    conv_kernel<<<ROWS, DIMC, 0, stream>>>(idxg, NLK, vertices, fo, D_l, pre);
    bn_relu_kernel<<<DIMC, 256, 0, stream>>>(pre, g_l, be_l, out, NOUT, 0);

    // 3) global branch, stage 0
    gemm_bias_wmma<<<gemm_blocks, 256, 0, stream>>>(feats, W_g0, b_g0, fo);
    conv_kernel<<<ROWS, DIMC, 0, stream>>>(idxg, NGK, vertices, fo, D_g0, pre);
    bn_relu_kernel<<<DIMC, 256, 0, stream>>>(pre, g_g0, be_g0, fmg, DIMC, 0);

    // 4) global branch, stage 1 -> out channels [128..255]
    gemm_bias_wmma<<<gemm_blocks, 256, 0, stream>>>(fmg, W_g1, b_g1, fo);
    conv_kernel<<<ROWS, DIMC, 0, stream>>>(idxg, NGK, vertices, fo, D_g1, pre);
    bn_relu_kernel<<<DIMC, 256, 0, stream>>>(pre, g_g1, be_g1, out, NOUT, DIMC);
}